// FFTSelector_8237747274373
// MI455X (gfx1250) — compile-verified
//
#include <hip/hip_runtime.h>

// ---------------------------------------------------------------------------
// FFTSelector for MI455X (gfx1250, wave32).
//
// Identity: sum_h irfft(qf*conj(kf))[h] = Re[(qf*conj(kf))[0]]
//                                       = (sum_h q[h]) * (sum_h k[h]).
// =>  score[i,j] = 1/(B*N*H) * sum_{b,n} Qs[b,i,n]*Ks[b,j,n]
//     Qs[b,t,n]  = x[b,t,n,:] . rowsum_h(Wq) + sum(bq)     (rank-1 collapse)
//
// Pass 0: rowsum of Wq/Wk + bias sums (trivial).
// Pass 1: stream cat_x once (50.3 MB -> ~2.2 us at 23.3 TB/s = the roofline),
//         emit Qs/Ks as [16384 samples][16] rows (T=12 zero-padded to 16).
// Pass 2: Gram S = Q^T K via V_WMMA_F32_16X16X4_F32, 64 blocks x 8 waves,
//         deterministic partial-matrix reduction (no float atomics).
// Pass 3: scale by 1/(B*N*H), per-row top-4 (sorted, first-index tie-break),
//         scores -> d_out[0..47] (f32), indices -> d_out[48..95] (int32 bits).
//
// Workspace (floats): [0..63] wq_sum | [64..127] wk_sum | [128..129] biases |
//   [256 ..) Qs (262144) | Ks (262144) | partials (64*256).  ~2.2 MB total.
// ---------------------------------------------------------------------------

typedef __attribute__((ext_vector_type(2))) float v2f;
typedef __attribute__((ext_vector_type(8))) float v8f;

#define BDIM 32
#define TDIM 12
#define TP   16
#define NDIM 512
#define FDIM 64
#define HDIM 64
#define STOT (BDIM * NDIM)               /* 16384 samples (b,n)            */
#define TAU  4
#define SCALE (1.0f / (float)(BDIM * NDIM * HDIM)) /* 1/1048576            */

#define WS_WQ 0
#define WS_WK 64
#define WS_B  128                        /* [128]=bq_sum, [129]=bk_sum     */
#define WS_QS 256
#define WS_KS (WS_QS + STOT * TP)
#define WS_PT (WS_KS + STOT * TP)        /* partial 16x16 Gram matrices    */
#define NBLK2 64                         /* Gram blocks                    */
#define SPB   (STOT / NBLK2)             /* 256 samples per block          */
#define SPW   (SPB / 8)                  /* 32 samples per wave            */

// ---- Pass 0: weight row-sums + bias sums ----------------------------------
__global__ __launch_bounds__(128) void fsel_sums_kernel(
    const float* __restrict__ Wq, const float* __restrict__ bq,
    const float* __restrict__ Wk, const float* __restrict__ bk,
    float* __restrict__ ws) {
  const int tid = threadIdx.x;
  if (tid < 64) {
    float s = 0.f;
#pragma unroll
    for (int h = 0; h < HDIM; ++h) s += Wq[tid * HDIM + h];
    ws[WS_WQ + tid] = s;
  } else {
    const int f = tid - 64;
    float s = 0.f;
#pragma unroll
    for (int h = 0; h < HDIM; ++h) s += Wk[f * HDIM + h];
    ws[WS_WK + f] = s;
  }
  if (tid == 0) {
    float s = 0.f;
#pragma unroll
    for (int h = 0; h < HDIM; ++h) s += bq[h];
    ws[WS_B + 0] = s;
  }
  if (tid == 1) {
    float s = 0.f;
#pragma unroll
    for (int h = 0; h < HDIM; ++h) s += bk[h];
    ws[WS_B + 1] = s;
  }
}

// ---- Pass 1: streaming Q/K hidden-sums (memory-bound roofline pass) -------
// One thread per (sample, padded-t). gid = s*16 + t  ->  stores coalesced.
__global__ __launch_bounds__(256) void fsel_qk_kernel(
    const float* __restrict__ x, const float* __restrict__ ws,
    float* __restrict__ Qs, float* __restrict__ Ks) {
  __shared__ float4 sq[16], sk[16];
  __shared__ float  sb[2];
  const int tid = threadIdx.x;
  if (tid < 16)       sq[tid]      = ((const float4*)(ws + WS_WQ))[tid];
  else if (tid < 32)  sk[tid - 16] = ((const float4*)(ws + WS_WK))[tid - 16];
  else if (tid < 34)  sb[tid - 32] = ws[WS_B + tid - 32];
  __syncthreads();

  const int gid = blockIdx.x * 256 + tid;   // 0 .. STOT*TP-1
  const int s = gid >> 4;
  const int t = gid & 15;
  float q = 0.f, k = 0.f;
  if (t < TDIM) {
    const int b = s >> 9;                   // / NDIM
    const int n = s & (NDIM - 1);
    const float4* __restrict__ xv =
        (const float4*)(x + (size_t)((b * TDIM + t) * NDIM + n) * FDIM);
    float4 aq = {0.f, 0.f, 0.f, 0.f}, ak = {0.f, 0.f, 0.f, 0.f};
#pragma unroll
    for (int i = 0; i < 16; ++i) {
      const float4 xx = xv[i];
      const float4 wq = sq[i], wk = sk[i];
      aq.x += xx.x * wq.x; aq.y += xx.y * wq.y;
      aq.z += xx.z * wq.z; aq.w += xx.w * wq.w;
      ak.x += xx.x * wk.x; ak.y += xx.y * wk.y;
      ak.z += xx.z * wk.z; ak.w += xx.w * wk.w;
    }
    q = (aq.x + aq.y) + (aq.z + aq.w) + sb[0];
    k = (ak.x + ak.y) + (ak.z + ak.w) + sb[1];
  }
  Qs[gid] = q;   // zero padding for t >= 12 keeps the Gram exact
  Ks[gid] = k;
}

// ---- Pass 2: Gram S = Q^T K via V_WMMA_F32_16X16X4_F32 --------------------
// A (16x4 f32): lanes 0-15 row M=lane, v0/v1 = K0/K1; lanes 16-31 = K2/K3.
// B (4x16 f32): mirrored — v0/v1 rows K0/K1 in lanes 0-15, K2/K3 in 16-31.
// With Qs/Ks stored as [sample][16], both operand loads are coalesced.
__global__ __launch_bounds__(256) void fsel_gram_kernel(
    const float* __restrict__ Qs, const float* __restrict__ Ks,
    float* __restrict__ part) {
  __shared__ float red[8][256];
  const int tid  = threadIdx.x;
  const int w    = tid >> 5;
  const int lane = tid & 31;
  const int half = lane >> 4;     // 0: K0/K1 rows, 1: K2/K3 rows
  const int col  = lane & 15;     // time index (M for A, N for B)

  v8f c = {};
  const int s_begin = blockIdx.x * SPB + w * SPW;
#pragma unroll 4
  for (int s0 = s_begin; s0 < s_begin + SPW; s0 += 4) {
    const int r = (s0 + 2 * half) * TP + col;
    v2f a, b;
    a.x = Qs[r];      a.y = Qs[r + TP];
    b.x = Ks[r];      b.y = Ks[r + TP];
    // D = A x B + C : 16x16 f32 accumulator, K=4 samples per issue
    c = __builtin_amdgcn_wmma_f32_16x16x4_f32(
        /*neg_a=*/false, a, /*neg_b=*/false, b,
        /*c_mod=*/(short)0, c, /*reuse_a=*/false, /*reuse_b=*/false);
  }

#pragma unroll
  for (int v = 0; v < 8; ++v) red[w][v * 32 + lane] = c[v];
  __syncthreads();

  float sum = 0.f;                 // element e = tid of the 16x16 tile
#pragma unroll
  for (int ww = 0; ww < 8; ++ww) sum += red[ww][tid];
  part[blockIdx.x * 256 + tid] = sum;
}

// ---- Pass 3: reduce partials, scale, per-row top-4 ------------------------
__global__ __launch_bounds__(256) void fsel_topk_kernel(
    const float* __restrict__ part, float* __restrict__ out) {
  __shared__ float score[16][16];
  const int tid = threadIdx.x;
  float sum = 0.f;
#pragma unroll
  for (int p = 0; p < NBLK2; ++p) sum += part[p * 256 + tid];
  // C/D layout: VGPR v, lanes 0-15 -> M=v, lanes 16-31 -> M=v+8; N=lane&15
  const int v = tid >> 5, L = tid & 31;
  score[v + 8 * (L >> 4)][L & 15] = sum * SCALE;
  __syncthreads();

  if (tid < TDIM) {
    float row[TDIM];
#pragma unroll
    for (int j = 0; j < TDIM; ++j) row[j] = score[tid][j];
#pragma unroll
    for (int r = 0; r < TAU; ++r) {       // sorted desc, lowest-index ties
      float best = -__builtin_inff();
      int bj = 0;
#pragma unroll
      for (int j = 0; j < TDIM; ++j)
        if (row[j] > best) { best = row[j]; bj = j; }
      out[tid * TAU + r] = best;                                // scores f32
      ((int*)out)[TDIM * TAU + tid * TAU + r] = bj;             // indices i32
      row[bj] = -__builtin_inff();
    }
  }
}

extern "C" void kernel_launch(void* const* d_in, const int* in_sizes, int n_in,
                              void* d_out, int out_size, void* d_ws, size_t ws_size,
                              hipStream_t stream) {
  const float* cat_x = (const float*)d_in[0];
  const float* Wq    = (const float*)d_in[1];
  const float* bq    = (const float*)d_in[2];
  const float* Wk    = (const float*)d_in[3];
  const float* bk    = (const float*)d_in[4];
  // d_in[5] is tau (device scalar); reference fixes tau = 4.

  float* ws   = (float*)d_ws;          // needs ~2.2 MB (see layout above)
  float* Qs   = ws + WS_QS;
  float* Ks   = ws + WS_KS;
  float* part = ws + WS_PT;
  float* out  = (float*)d_out;         // 48 f32 scores + 48 i32 indices

  fsel_sums_kernel<<<1, 128, 0, stream>>>(Wq, bq, Wk, bk, ws);
  fsel_qk_kernel<<<(STOT * TP) / 256, 256, 0, stream>>>(cat_x, ws, Qs, Ks);
  fsel_gram_kernel<<<NBLK2, 256, 0, stream>>>(Qs, Ks, part);
  fsel_topk_kernel<<<1, 256, 0, stream>>>(part, out);
}